// Atten_55688545960062
// MI455X (gfx1250) — compile-verified
//
#include <hip/hip_runtime.h>
#include <math.h>

// Problem constants (reference: B=64, T=256, R=49 regions, H=1024 hidden)
#define B_  64
#define T_  256
#define R_  49
#define H_  1024

typedef float v2f __attribute__((ext_vector_type(2)));
typedef float v8f __attribute__((ext_vector_type(8)));

// D = A(16x4 f32) * B(4x16 f32) + C(16x16 f32), wave32 WMMA.
// 8-arg pattern: (neg_a, A, neg_b, B, c_mod, C, reuse_a, reuse_b)
__device__ __forceinline__ v8f wmma_f32(v2f a, v2f b, v8f c) {
  return __builtin_amdgcn_wmma_f32_16x16x4_f32(
      false, a, false, b, (short)0, c, false, false);
}

// Saturating tanh via v_exp_f32 (transcendental pipe), deterministic.
__device__ __forceinline__ float fast_tanh(float x) {
  x = fminf(fmaxf(x, -9.f), 9.f);
  float e = __expf(2.f * x);
  return (e - 1.f) / (e + 1.f);
}

// ---------------------------------------------------------------------------
// Kernel 1: cv[b,r,k] = sum_h V[b,r,h] * Wv[k,h]   ([49x1024]x[1024x49] per b)
// Block = 256 threads (8 waves). M,N padded 49->64 (4x4 tiles of 16x16).
// Wave w owns tiles (mt = w/2, nt = (w%2)*2 + {0,1}).
// ---------------------------------------------------------------------------
__global__ void cv_kernel(const float* __restrict__ V,
                          const float* __restrict__ Wv,
                          float* __restrict__ cv) {
  __shared__ float Ash[64][66];   // [m][h] chunk of V[b]
  __shared__ float Bsh[64][72];   // [h][n] chunk of Wv^T (stride 72: 2*72%64==16)
  const int b    = blockIdx.x;
  const int tid  = threadIdx.x;
  const int wave = tid >> 5, lane = tid & 31;
  const int lr = lane & 15, lh = lane >> 4;
  const int mt = wave >> 1, ntB = (wave & 1) * 2;
  v8f acc0 = {}, acc1 = {};

  for (int hc = 0; hc < H_; hc += 64) {
    #pragma unroll
    for (int i = 0; i < 16; ++i) {              // 64x64 A chunk, coalesced on h
      int idx = tid + i * 256;
      int h = idx & 63, m = idx >> 6;
      Ash[m][h] = (m < R_) ? V[((size_t)b * R_ + m) * H_ + hc + h] : 0.f;
    }
    #pragma unroll
    for (int i = 0; i < 16; ++i) {              // 64x64 B chunk (transposed)
      int idx = tid + i * 256;
      int h = idx & 63, n = idx >> 6;
      Bsh[h][n] = (n < R_) ? Wv[(size_t)n * H_ + hc + h] : 0.f;
    }
    __syncthreads();
    #pragma unroll
    for (int ks = 0; ks < 64; ks += 4) {
      int k0 = ks + lh * 2;                     // half-wave K pair select
      v2f a;  a.x  = Ash[mt * 16 + lr][k0];
              a.y  = Ash[mt * 16 + lr][k0 + 1];
      v2f b0; b0.x = Bsh[k0][ntB * 16 + lr];
              b0.y = Bsh[k0 + 1][ntB * 16 + lr];
      v2f b1; b1.x = Bsh[k0][(ntB + 1) * 16 + lr];
              b1.y = Bsh[k0 + 1][(ntB + 1) * 16 + lr];
      acc0 = wmma_f32(a, b0, acc0);
      acc1 = wmma_f32(a, b1, acc1);
    }
    __syncthreads();
  }
  #pragma unroll
  for (int v = 0; v < 8; ++v) {                 // C layout: row = v + half*8
    int row = mt * 16 + v + lh * 8;
    int c0 = ntB * 16 + lr, c1 = (ntB + 1) * 16 + lr;
    if (row < R_) {
      if (c0 < R_) cv[((size_t)b * R_ + row) * R_ + c0] = acc0[v];
      if (c1 < R_) cv[((size_t)b * R_ + row) * R_ + c1] = acc1[v];
    }
  }
}

// ---------------------------------------------------------------------------
// Kernel 2: cg[m,k] = sum_h h_t[m,h] * Wg[k,h],  m = b*T+t (16384 rows)
// Block covers 128 rows (wave w -> 16-row tile), N padded 49->64 (4 tiles).
// ---------------------------------------------------------------------------
__global__ void cg_kernel(const float* __restrict__ hT,
                          const float* __restrict__ Wg,
                          float* __restrict__ cg) {
  __shared__ float Ash[128][66];
  __shared__ float Bsh[64][72];
  const int mBase = blockIdx.x * 128;
  const int tid   = threadIdx.x;
  const int wave  = tid >> 5, lane = tid & 31;
  const int lr = lane & 15, lh = lane >> 4;
  v8f acc[4] = {{}, {}, {}, {}};

  for (int hc = 0; hc < H_; hc += 64) {
    #pragma unroll
    for (int i = 0; i < 32; ++i) {              // 128x64 A chunk
      int idx = tid + i * 256;
      int h = idx & 63, m = idx >> 6;
      Ash[m][h] = hT[(size_t)(mBase + m) * H_ + hc + h];
    }
    #pragma unroll
    for (int i = 0; i < 16; ++i) {              // 64x64 Wg chunk (transposed)
      int idx = tid + i * 256;
      int h = idx & 63, n = idx >> 6;
      Bsh[h][n] = (n < R_) ? Wg[(size_t)n * H_ + hc + h] : 0.f;
    }
    if (hc + 64 < H_)                           // gfx1250 global_prefetch_b8
      __builtin_prefetch(&hT[(size_t)(mBase + (tid >> 1)) * H_ + hc + 64 +
                             (tid & 1) * 32], 0, 1);
    __syncthreads();
    #pragma unroll 4
    for (int ks = 0; ks < 64; ks += 4) {
      int k0 = ks + lh * 2;
      v2f a; a.x = Ash[wave * 16 + lr][k0];
             a.y = Ash[wave * 16 + lr][k0 + 1];
      #pragma unroll
      for (int nt = 0; nt < 4; ++nt) {
        v2f bb; bb.x = Bsh[k0][nt * 16 + lr];
                bb.y = Bsh[k0 + 1][nt * 16 + lr];
        acc[nt] = wmma_f32(a, bb, acc[nt]);
      }
    }
    __syncthreads();
  }
  #pragma unroll
  for (int nt = 0; nt < 4; ++nt) {
    #pragma unroll
    for (int v = 0; v < 8; ++v) {
      int row = mBase + wave * 16 + v + lh * 8;
      int col = nt * 16 + lr;
      if (col < R_) cg[(size_t)row * R_ + col] = acc[nt][v];
    }
  }
}

// ---------------------------------------------------------------------------
// Kernel 3: z[r] = sum_k tanh(cv[b,r,k]+cg[b,t,k])*Wh[k]; alpha = softmax(z).
// One wave per (b,t); 8 t per block. cv[b] cached in LDS (stride 53 so lanes
// L and L+16 hit disjoint banks). Wave32 shfl_xor reductions.
// ---------------------------------------------------------------------------
__global__ void attn_kernel(const float* __restrict__ cv,
                            const float* __restrict__ cg,
                            const float* __restrict__ Wh,
                            float* __restrict__ alpha) {
  __shared__ float cvsh[R_][53];
  __shared__ float cgsh[8][53];
  __shared__ float whsh[56];
  const int b   = blockIdx.y;
  const int t0  = blockIdx.x * 8;
  const int tid = threadIdx.x;
  const int wave = tid >> 5, lane = tid & 31;

  for (int i = tid; i < R_ * R_; i += 256)
    cvsh[i / R_][i % R_] = cv[(size_t)b * R_ * R_ + i];
  for (int i = tid; i < 8 * R_; i += 256)
    cgsh[i / R_][i % R_] = cg[(size_t)(b * T_ + t0) * R_ + i];
  for (int i = tid; i < R_; i += 256) whsh[i] = Wh[i];
  __syncthreads();

  const int r0 = lane;          // always < 49
  const int r1 = lane + 32;     // valid for lanes 0..16
  float z0 = 0.f, z1 = 0.f;
  for (int k = 0; k < R_; ++k) {
    float g  = cgsh[wave][k];
    float wk = whsh[k];
    z0 += fast_tanh(cvsh[r0][k] + g) * wk;
    if (r1 < R_) z1 += fast_tanh(cvsh[r1][k] + g) * wk;
  }
  // softmax over 49 values spread across one wave32
  float m = (r1 < R_) ? fmaxf(z0, z1) : z0;
  for (int off = 16; off > 0; off >>= 1) m = fmaxf(m, __shfl_xor(m, off, 32));
  float e0 = __expf(z0 - m);
  float e1 = (r1 < R_) ? __expf(z1 - m) : 0.f;
  float s = e0 + e1;
  for (int off = 16; off > 0; off >>= 1) s += __shfl_xor(s, off, 32);
  float inv = __builtin_amdgcn_rcpf(s);
  const size_t base = (size_t)(b * T_ + t0 + wave) * R_;
  alpha[base + r0] = e0 * inv;
  if (r1 < R_) alpha[base + r1] = e1 * inv;
}

// ---------------------------------------------------------------------------
// Kernel 4: c_t[b,t,:] = sum_r alpha[b,t,r] * V[b,r,:]
// Per-batch GEMM [256x49]x[49x1024] via WMMA, K padded 49->52 (13 k-steps).
// Block: 16 t-rows x 128 cols (wave w -> 16-col tile).
// ---------------------------------------------------------------------------
__global__ void ct_kernel(const float* __restrict__ alpha,
                          const float* __restrict__ V,
                          float* __restrict__ ct) {
  __shared__ float Ash[16][53];    // alpha rows, K padded with zeros
  __shared__ float Bsh[52][136];   // V[b] col panel (stride 136: 2*136%64==16)
  const int mTile = blockIdx.x;
  const int nBase = blockIdx.y * 128;
  const int b     = blockIdx.z;
  const int tid   = threadIdx.x;
  const int wave  = tid >> 5, lane = tid & 31;
  const int lr = lane & 15, lh = lane >> 4;

  for (int i = tid; i < 16 * 53; i += 256) {
    int tl = i / 53, k = i % 53;
    Ash[tl][k] = (k < R_)
        ? alpha[(size_t)(b * T_ + mTile * 16 + tl) * R_ + k] : 0.f;
  }
  for (int i = tid; i < 52 * 128; i += 256) {
    int k = i >> 7, n = i & 127;
    Bsh[k][n] = (k < R_) ? V[((size_t)b * R_ + k) * H_ + nBase + n] : 0.f;
  }
  __syncthreads();

  v8f acc = {};
  #pragma unroll
  for (int ks = 0; ks < 52; ks += 4) {
    int k0 = ks + lh * 2;
    v2f a;  a.x  = Ash[lr][k0];
            a.y  = Ash[lr][k0 + 1];
    v2f bb; bb.x = Bsh[k0][wave * 16 + lr];
            bb.y = Bsh[k0 + 1][wave * 16 + lr];
    acc = wmma_f32(a, bb, acc);
  }
  #pragma unroll
  for (int v = 0; v < 8; ++v) {
    int t   = mTile * 16 + v + lh * 8;
    int col = nBase + wave * 16 + lr;
    ct[(size_t)(b * T_ + t) * H_ + col] = acc[v];
  }
}

// ---------------------------------------------------------------------------
extern "C" void kernel_launch(void* const* d_in, const int* in_sizes, int n_in,
                              void* d_out, int out_size, void* d_ws, size_t ws_size,
                              hipStream_t stream) {
  (void)in_sizes; (void)n_in; (void)out_size; (void)ws_size;
  const float* V  = (const float*)d_in[0];   // [B,R,H]
  const float* hT = (const float*)d_in[1];   // [B,T,H]
  const float* Wv = (const float*)d_in[2];   // [R,H]
  const float* Wg = (const float*)d_in[3];   // [R,H]
  const float* Wh = (const float*)d_in[4];   // [1,R]

  float* ct    = (float*)d_out;                      // output 0: [B,T,H]
  float* alpha = ct + (size_t)B_ * T_ * H_;          // output 1: [B,T,R]

  float* wcv = (float*)d_ws;                         // [B,R,R]   ~0.6 MB
  float* wcg = wcv + (size_t)B_ * R_ * R_;           // [B*T,R]   ~3.2 MB

  cv_kernel  <<<dim3(B_),                    dim3(256), 0, stream>>>(V,  Wv, wcv);
  cg_kernel  <<<dim3((B_ * T_) / 128),       dim3(256), 0, stream>>>(hT, Wg, wcg);
  attn_kernel<<<dim3(T_ / 8, B_),            dim3(256), 0, stream>>>(wcv, wcg, Wh, alpha);
  ct_kernel  <<<dim3(T_ / 16, H_ / 128, B_), dim3(256), 0, stream>>>(alpha, V, ct);
}